// V2STransformer_34110630265653
// MI455X (gfx1250) — compile-verified
//
#include <hip/hip_runtime.h>

// Volume-to-slice affine resampling for MI455X (gfx1250, wave32).
//
// Shapes (fixed by reference): vol [8,160,160,96], trf [8,64,12], out [8,160,160,64].
//
// Roofline: ~131 MB HBM traffic @ 23.3 TB/s => ~5.6 us floor; 0.46 GFLOP is noise.
// => gather-bound kernel. Design:
//  * slices-on-lanes: gathers step ~6B along contiguous D axis (z = 1.5*s), stores
//    are contiguous in s => two 64B segments per store instruction.
//  * per-batch 3x4 matrices staged to LDS via CDNA5 async-to-LDS DMA (ASYNCcnt path).
//  * coordinate transform via V_WMMA_F32_16X16X4_F32: D[pixel][slice] =
//    [i, j, 1, 0] x [a'c0, a'c1, a'c2*z_s + a'c3, 0]^T ; 3 WMMAs yield x/y/z for a
//    16-pixel x 16-slice tile, landing per-lane exactly where the gather needs them.
//  * ALL gather/store indexing in 32-bit so the backend emits SADDR+voffset global
//    loads (confirmed in disasm: global_load_b32 v, v_off, s[0:1] inside s_clause).
//  * border clamps via v_med3_num_f32 directly (no canonicalize max).

#define B_  8
#define H_  160
#define W_  160
#define D_  96
#define S_  64
#define ST_RATIO 1.5f

typedef __attribute__((ext_vector_type(2))) float v2f;
typedef __attribute__((ext_vector_type(8))) float v8f;

__global__ __launch_bounds__(256) void v2s_wmma_kernel(
    const float* __restrict__ vol,
    const float* __restrict__ trf,
    float* __restrict__ out)
{
    __shared__ __align__(16) float Amat[S_ * 12];  // 3 KB: all 64 slice transforms of batch b

    const int tid  = threadIdx.x;
    const int b    = blockIdx.y;
    const int wave = tid >> 5;
    const int lane = tid & 31;
    const int half = lane >> 4;   // 0: D rows 0-7 / B rows K0,K1 ; 1: D rows 8-15 / B rows K2,K3
    const int nn   = lane & 15;   // slice column within tile / pixel row for A operand

    // ---- Stage trf[b] (768 floats = 192 x b128) into LDS via async DMA ----
    {
        const float* gsrc = trf + b * (S_ * 12);
        if (tid < 192) {
            unsigned long long ga = (unsigned long long)(const void*)(gsrc + tid * 4);
            unsigned la = (unsigned)(size_t)(&Amat[0]) + (unsigned)tid * 16u;
            asm volatile("global_load_async_to_lds_b128 %0, %1, off"
                         :: "v"(la), "v"(ga) : "memory");
        }
        asm volatile("s_wait_asynccnt 0x0" ::: "memory");
    }
    __syncthreads();

    // ---- This wave's 16 pixels: 16 consecutive w at fixed h (W=160 is 16-divisible) ----
    const int p0 = blockIdx.x * 128 + wave * 16;  // flattened h*W + w, 16-aligned
    const int h  = p0 / W_;
    const int w0 = p0 % W_;

    // WMMA A operand (16x4 f32): row m = [i, j_m, 1, 0].
    // Layout: lanes 0-15 hold K=0,1 of row=lane; lanes 16-31 hold K=2,3 of row=lane-16.
    v2f aop;
    if (half == 0) { aop.x = (float)h; aop.y = (float)(w0 + nn); }
    else           { aop.x = 1.0f;     aop.y = 0.0f; }

    // Uniform 32-bit-indexable bases (per-batch volume = 9.8 MB, out = 52 MB).
    const char* volc = (const char*)(vol + (size_t)b * (H_ * W_ * D_));

    for (int st = 0; st < 4; ++st) {
        const int s  = st * 16 + nn;              // slice this lane owns (B column / D column)
        const float* As = &Amat[s * 12];
        const float zz  = ST_RATIO * (float)s;

        // Three WMMAs: one per coordinate component.
        v8f coord[3];
#pragma unroll
        for (int c = 0; c < 3; ++c) {
            v2f bop;
            if (half == 0) {   // B rows K=0 (a'c0) and K=1 (a'c1), column = slice nn
                bop.x = As[c * 4 + 0] + (c == 0 ? 1.0f : 0.0f);
                bop.y = As[c * 4 + 1] + (c == 1 ? 1.0f : 0.0f);
            } else {           // B rows K=2 (z-folded translation) and K=3 (pad 0)
                float azc = As[c * 4 + 2] + (c == 2 ? 1.0f : 0.0f);
                bop.x = fmaf(azc, zz, As[c * 4 + 3]);
                bop.y = 0.0f;
            }
            v8f cz = {};
            coord[c] = __builtin_amdgcn_wmma_f32_16x16x4_f32(
                false, aop, false, bop, (short)0, cz, false, false);
        }

        // D layout: lane(half,nn) element k -> pixel m = k + 8*half, slice column nn.
        // 32-bit element index into out (max 13.1M, fits int).
        const int obase = (((b * H_ + h) * W_ + (w0 + 8 * half)) * S_) + s;

#pragma unroll
        for (int k = 0; k < 8; ++k) {
            // Single-instruction border clamps (v_med3_num_f32).
            float x = __builtin_amdgcn_fmed3f(coord[0][k], 0.0f, (float)(H_ - 1));
            float y = __builtin_amdgcn_fmed3f(coord[1][k], 0.0f, (float)(W_ - 1));
            float z = __builtin_amdgcn_fmed3f(coord[2][k], 0.0f, (float)(D_ - 1));
            float fx = floorf(x), fy = floorf(y), fz = floorf(z);
            float dx = x - fx,   dy = y - fy,   dz = z - fz;
            int ix0 = (int)fx,   iy0 = (int)fy, iz0 = (int)fz;
            int ix1 = min(ix0 + 1, H_ - 1);
            int iy1 = min(iy0 + 1, W_ - 1);
            int iz1 = min(iz0 + 1, D_ - 1);

            // Byte offsets, all 32-bit: row strides W*D*4 = 61440, D*4 = 384.
            const int bx0 = ix0 * (W_ * D_ * 4);
            const int bx1 = ix1 * (W_ * D_ * 4);
            const int by0 = iy0 * (D_ * 4);
            const int by1 = iy1 * (D_ * 4);
            const int bz0 = iz0 * 4;
            const int bz1 = iz1 * 4;
            const int r00 = bx0 + by0;
            const int r01 = bx0 + by1;
            const int r10 = bx1 + by0;
            const int r11 = bx1 + by1;

            float v000 = *(const float*)(volc + (r00 + bz0));
            float v001 = *(const float*)(volc + (r00 + bz1));
            float v010 = *(const float*)(volc + (r01 + bz0));
            float v011 = *(const float*)(volc + (r01 + bz1));
            float v100 = *(const float*)(volc + (r10 + bz0));
            float v101 = *(const float*)(volc + (r10 + bz1));
            float v110 = *(const float*)(volc + (r11 + bz0));
            float v111 = *(const float*)(volc + (r11 + bz1));

            float c00 = fmaf(dz, v001 - v000, v000);
            float c01 = fmaf(dz, v011 - v010, v010);
            float c10 = fmaf(dz, v101 - v100, v100);
            float c11 = fmaf(dz, v111 - v110, v110);
            float c0  = fmaf(dy, c01 - c00, c00);
            float c1  = fmaf(dy, c11 - c10, c10);
            out[obase + k * S_] = fmaf(dx, c1 - c0, c0);
        }
    }
}

extern "C" void kernel_launch(void* const* d_in, const int* in_sizes, int n_in,
                              void* d_out, int out_size, void* d_ws, size_t ws_size,
                              hipStream_t stream) {
    (void)in_sizes; (void)n_in; (void)out_size; (void)d_ws; (void)ws_size;
    const float* vol = (const float*)d_in[0];
    const float* trf = (const float*)d_in[1];
    float* out = (float*)d_out;

    // 8 waves/block, 16 pixels/wave => 128 pixels/block; H*W = 25600 => 200 blocks per batch.
    dim3 grid((H_ * W_) / 128, B_, 1);
    v2s_wmma_kernel<<<grid, dim3(256, 1, 1), 0, stream>>>(vol, trf, out);
}